// TGBase_10677288698289
// MI455X (gfx1250) — compile-verified
//
#include <hip/hip_runtime.h>
#include <stdint.h>

#define TGB 256

// ---------------- CDNA5 async global->LDS helpers (gfx1250) ----------------
// Low 32 bits of a flat pointer into the LDS aperture are the LDS byte address
// (ISA 10.2).
__device__ __forceinline__ unsigned tgb_lds_addr(const void* p) {
  return (unsigned)(uintptr_t)p;
}
__device__ __forceinline__ void tgb_async_b32(unsigned ldsOff, const void* g) {
  // GLOBAL_LOAD_ASYNC_TO_LDS_B32 (VGLOBAL op 96): per-lane 4B memory -> LDS, ASYNCcnt tracked
  asm volatile("global_load_async_to_lds_b32 %0, %1, off" :: "v"(ldsOff), "v"(g) : "memory");
}
#define TGB_WAIT_ASYNC(n) asm volatile("s_wait_asynccnt %0" :: "i"(n) : "memory")

// ---------------- init: zero accumulators ----------------
__global__ void tgb_init(float* sum, float* sumsq, unsigned* maxb, unsigned* minb,
                         float* clogc, float* degin, float* degout, int n9, int n) {
  int stride = gridDim.x * blockDim.x;
  for (int i = blockIdx.x * blockDim.x + threadIdx.x; i < n9; i += stride) {
    sum[i] = 0.f; sumsq[i] = 0.f; clogc[i] = 0.f;
    maxb[i] = 0u;              // all values >= 0, so 0 is a safe -inf surrogate
    minb[i] = 0x7f800000u;     // +inf
    if (i < n) { degin[i] = 0.f; degout[i] = 0.f; }
  }
}

// ---------------- pass 1: degrees ----------------
__global__ void tgb_degree(const int* __restrict__ src, const int* __restrict__ dst,
                           float* degin, float* degout, int E) {
  int stride = gridDim.x * blockDim.x;
  for (int e = blockIdx.x * blockDim.x + threadIdx.x; e < E; e += stride) {
    atomicAdd(&degout[src[e]], 1.f);
    atomicAdd(&degin[dst[e]], 1.f);
  }
}

// ---------------- pass 2: per-edge stat scatter (async-LDS staged) ----------------
__device__ __forceinline__ void tgb_contrib(float* sum, float* sumsq,
                                            unsigned* maxb, unsigned* minb,
                                            int base, float v) {
  atomicAdd(sum + base, v);
  atomicAdd(sumsq + base, v * v);
  unsigned b = __float_as_uint(v);   // v >= 0 -> uint order == float order
  atomicMax(maxb + base, b);
  atomicMin(minb + base, b);
}

__global__ __launch_bounds__(TGB) void tgb_stats(
    const int* __restrict__ src, const int* __restrict__ dst,
    const float* __restrict__ w, const float* __restrict__ t,
    const float* __restrict__ degin, const float* __restrict__ degout,
    float* sum, float* sumsq, unsigned* maxb, unsigned* minb,
    int E, int n, int numTiles) {
  __shared__ int   sh_s[2][TGB];
  __shared__ int   sh_d[2][TGB];
  __shared__ float sh_w[2][TGB];
  __shared__ float sh_t[2][TGB];

  const int tid = threadIdx.x;
  const int gstride = gridDim.x;
  int tile = blockIdx.x;

  if (tile < numTiles) {   // prologue: stage first tile into buffer 0
    int idx = min(tile * TGB + tid, E - 1);
    tgb_async_b32(tgb_lds_addr(&sh_s[0][tid]), src + idx);
    tgb_async_b32(tgb_lds_addr(&sh_d[0][tid]), dst + idx);
    tgb_async_b32(tgb_lds_addr(&sh_w[0][tid]), w + idx);
    tgb_async_b32(tgb_lds_addr(&sh_t[0][tid]), t + idx);
  }
  int pb = 0;
  for (; tile < numTiles; tile += gstride, pb ^= 1) {
    int nxt = tile + gstride;
    if (nxt < numTiles) {  // prefetch next tile, then retire current (in-order completion)
      int idx = min(nxt * TGB + tid, E - 1);
      tgb_async_b32(tgb_lds_addr(&sh_s[pb ^ 1][tid]), src + idx);
      tgb_async_b32(tgb_lds_addr(&sh_d[pb ^ 1][tid]), dst + idx);
      tgb_async_b32(tgb_lds_addr(&sh_w[pb ^ 1][tid]), w + idx);
      tgb_async_b32(tgb_lds_addr(&sh_t[pb ^ 1][tid]), t + idx);
      TGB_WAIT_ASYNC(4);
    } else {
      TGB_WAIT_ASYNC(0);
    }
    int e = tile * TGB + tid;
    if (e < E) {
      int s = sh_s[pb][tid], d = sh_d[pb][tid];
      float wv = sh_w[pb][tid], tv = sh_t[pb][tid];
      float ins  = degin[s],  outs = degout[s];
      float ind  = degin[d],  outd = degout[d];
      float tots = ins + outs, totd = ind + outd;
      // struct variants: A:(seg=d, in[s])  B:(seg=s, out[d])  C:(s, tot[d]) + (d, tot[s])
      tgb_contrib(sum, sumsq, maxb, minb, 0 * n + d, ins);
      tgb_contrib(sum, sumsq, maxb, minb, 1 * n + s, outd);
      tgb_contrib(sum, sumsq, maxb, minb, 2 * n + s, totd);
      tgb_contrib(sum, sumsq, maxb, minb, 2 * n + d, tots);
      // weights
      tgb_contrib(sum, sumsq, maxb, minb, 3 * n + d, wv);
      tgb_contrib(sum, sumsq, maxb, minb, 4 * n + s, wv);
      tgb_contrib(sum, sumsq, maxb, minb, 5 * n + s, wv);
      tgb_contrib(sum, sumsq, maxb, minb, 5 * n + d, wv);
      // timestamps
      tgb_contrib(sum, sumsq, maxb, minb, 6 * n + d, tv);
      tgb_contrib(sum, sumsq, maxb, minb, 7 * n + s, tv);
      tgb_contrib(sum, sumsq, maxb, minb, 8 * n + s, tv);
      tgb_contrib(sum, sumsq, maxb, minb, 8 * n + d, tv);
    }
  }
}

// ---------------- entropy: hash-based exact run counts ----------------
__global__ void tgb_hclear(unsigned long long* hkey, unsigned* hcnt, unsigned HS) {
  unsigned stride = gridDim.x * blockDim.x;
  for (unsigned i = blockIdx.x * blockDim.x + threadIdx.x; i < HS; i += stride) {
    hkey[i] = ~0ull; hcnt[i] = 0u;
  }
}

__device__ __forceinline__ void tgb_hinsert(unsigned long long* hkey, unsigned* hcnt,
                                            float* clogc, int n, int pass, unsigned hmask,
                                            int seg, float val) {
  unsigned long long key =
      ((unsigned long long)(unsigned)seg << 32) | (unsigned long long)__float_as_uint(val);
  unsigned long long z = key + 0x9e3779b97f4a7c15ull;          // splitmix64 mix
  z = (z ^ (z >> 30)) * 0xbf58476d1ce4e5b9ull;
  z = (z ^ (z >> 27)) * 0x94d049bb133111ebull;
  z ^= z >> 31;
  unsigned slot = (unsigned)z & hmask;
  for (int probe = 0; probe < 4096; ++probe) {
    unsigned long long old = atomicCAS(&hkey[slot], ~0ull, key);
    if (old == ~0ull || old == key) {
      unsigned j = atomicAdd(&hcnt[slot], 1u) + 1u;  // rank among duplicates
      if (j > 1u) {
        float fj = (float)j, fk = (float)(j - 1u);
        atomicAdd(&clogc[pass * n + seg], fj * log2f(fj) - fk * log2f(fk));
      }
      return;
    }
    slot = (slot + 1u) & hmask;
  }
}

__global__ void tgb_entropy(const int* __restrict__ src, const int* __restrict__ dst,
                            const float* __restrict__ w, const float* __restrict__ t,
                            const float* __restrict__ degin, const float* __restrict__ degout,
                            unsigned long long* hkey, unsigned* hcnt, float* clogc,
                            int E, int n, int pass, unsigned hmask) {
  int stride = gridDim.x * blockDim.x;
  int vt = pass / 3, var = pass % 3;   // vt: 0=struct 1=weights 2=ts
  for (int e = blockIdx.x * blockDim.x + threadIdx.x; e < E; e += stride) {
    int s = src[e], d = dst[e];
    if (vt == 0) {
      if (var == 0)      tgb_hinsert(hkey, hcnt, clogc, n, pass, hmask, d, degin[s]);
      else if (var == 1) tgb_hinsert(hkey, hcnt, clogc, n, pass, hmask, s, degout[d]);
      else {
        tgb_hinsert(hkey, hcnt, clogc, n, pass, hmask, s, degin[d] + degout[d]);
        tgb_hinsert(hkey, hcnt, clogc, n, pass, hmask, d, degin[s] + degout[s]);
      }
    } else {
      float v = (vt == 1) ? w[e] : t[e];
      if (var == 0)      tgb_hinsert(hkey, hcnt, clogc, n, pass, hmask, d, v);
      else if (var == 1) tgb_hinsert(hkey, hcnt, clogc, n, pass, hmask, s, v);
      else {
        tgb_hinsert(hkey, hcnt, clogc, n, pass, hmask, s, v);
        tgb_hinsert(hkey, hcnt, clogc, n, pass, hmask, d, v);
      }
    }
  }
}

// ---------------- finalize: 57 columns per node ----------------
__global__ void tgb_final(const float* __restrict__ degin, const float* __restrict__ degout,
                          const float* __restrict__ sum, const float* __restrict__ sumsq,
                          const unsigned* __restrict__ maxb, const unsigned* __restrict__ minb,
                          const float* __restrict__ clogc, float* __restrict__ out, int n) {
  int stride = gridDim.x * blockDim.x;
  for (int nid = blockIdx.x * blockDim.x + threadIdx.x; nid < n; nid += stride) {
    float din = degin[nid], dout = degout[nid], tot = din + dout;
    float* row = out + (size_t)nid * 57;
    row[0] = din; row[1] = dout; row[2] = tot;
    for (int v = 0; v < 9; ++v) {
      float cnt = (v % 3 == 0) ? din : (v % 3 == 1) ? dout : tot;
      int b = v * n + nid;
      float s = sum[b], ss = sumsq[b];
      float mean = s / fmaxf(cnt, 1.f);
      float mx = (cnt > 0.f) ? __uint_as_float(maxb[b]) : 0.f;
      float mn = (cnt > 0.f) ? __uint_as_float(minb[b]) : 0.f;
      float sq = fmaxf(ss - s * mean, 0.f);            // == sum((v-mean)^2)
      float sd = sqrtf(sq / fmaxf(cnt - 1.f, 1.f));
      float ent = (cnt > 0.f) ? (log2f(cnt) - clogc[b] / cnt) : 0.f;
      int c = 3 + v * 6;
      row[c] = s; row[c + 1] = mean; row[c + 2] = mx;
      row[c + 3] = mn; row[c + 4] = sd; row[c + 5] = ent;
    }
  }
}

// ---------------- host ----------------
extern "C" void kernel_launch(void* const* d_in, const int* in_sizes, int n_in,
                              void* d_out, int out_size, void* d_ws, size_t ws_size,
                              hipStream_t stream) {
  const int*   ei = (const int*)d_in[0];
  const float* w  = (const float*)d_in[1];
  const float* t  = (const float*)d_in[2];
  const int E = in_sizes[1];
  const int n = out_size / 57;            // num_nodes (avoid reading device scalar)
  const int* src = ei;
  const int* dst = ei + E;

  // ---- workspace layout: [hkey | hcnt | degin | degout | sum | sumsq | maxb | minb | clogc]
  size_t fixedBytes = (size_t)(2 + 5 * 9) * (size_t)n * 4;
  size_t avail = (ws_size > fixedBytes) ? ws_size - fixedBytes : 0;
  size_t HS = 0;
  if (avail >= (size_t)1024 * 12) {
    HS = 1024;
    while ((HS << 1) * 12 <= avail && HS < ((size_t)1 << 24)) HS <<= 1;
  }
  char* p = (char*)d_ws;
  unsigned long long* hkey = (unsigned long long*)p;  p += HS * 8;
  unsigned* hcnt = (unsigned*)p;                      p += HS * 4;
  float* degin  = (float*)p;  p += (size_t)n * 4;
  float* degout = (float*)p;  p += (size_t)n * 4;
  float* sum    = (float*)p;  p += (size_t)9 * n * 4;
  float* sumsq  = (float*)p;  p += (size_t)9 * n * 4;
  unsigned* maxb = (unsigned*)p; p += (size_t)9 * n * 4;
  unsigned* minb = (unsigned*)p; p += (size_t)9 * n * 4;
  float* clogc  = (float*)p;

  const int n9 = 9 * n;
  int initBlocks = (n9 + TGB - 1) / TGB;
  tgb_init<<<initBlocks, TGB, 0, stream>>>(sum, sumsq, maxb, minb, clogc, degin, degout, n9, n);

  int degBlocks = (E + TGB - 1) / TGB;
  tgb_degree<<<degBlocks, TGB, 0, stream>>>(src, dst, degin, degout, E);

  int numTiles = (E + TGB - 1) / TGB;
  int statBlocks = numTiles < 4096 ? numTiles : 4096;
  tgb_stats<<<statBlocks, TGB, 0, stream>>>(src, dst, w, t, degin, degout,
                                            sum, sumsq, maxb, minb, E, n, numTiles);

  if (HS) {
    unsigned hmask = (unsigned)(HS - 1);
    int clrBlocks = (int)((HS + TGB - 1) / TGB);
    int entBlocks = degBlocks < 8192 ? degBlocks : 8192;
    for (int pass = 0; pass < 9; ++pass) {
      tgb_hclear<<<clrBlocks, TGB, 0, stream>>>(hkey, hcnt, (unsigned)HS);
      tgb_entropy<<<entBlocks, TGB, 0, stream>>>(src, dst, w, t, degin, degout,
                                                 hkey, hcnt, clogc, E, n, pass, hmask);
    }
  }

  int finBlocks = (n + TGB - 1) / TGB;
  tgb_final<<<finBlocks, TGB, 0, stream>>>(degin, degout, sum, sumsq, maxb, minb,
                                           clogc, (float*)d_out, n);
}